// MaternKernelTorch_35158602285242
// MI455X (gfx1250) — compile-verified
//
#include <hip/hip_runtime.h>
#include <hip/hip_bf16.h>

typedef __attribute__((ext_vector_type(2))) float v2f;
typedef __attribute__((ext_vector_type(8))) float v8f;

#define MAT_N 1024
#define MAT_D 32
#define NQ    192

#define LOG2E_F 1.44269504088896340736f
#define LN2_F   0.69314718055994530942f

// Workspace layout (floats):
//  [0 .. 1023]        row squared norms
//  [1024 .. 1791]     interleaved quad table: tab[2q] = -(cosh t_q - 1)*log2e
//                                             tab[2q+1] = w_q * cosh(nu t_q)
//  [1792] c_scale = sqrt(2*nu)/lengthscale
//  [1793] var
//  [1794] log_norm2 = ((1-nu)*ln2 - lgamma(nu)) * log2e
//  [1795] nu
#define WS_TAB   1024
#define WS_PARAM 1792

__global__ __launch_bounds__(256) void matern_sqn_kernel(const float* __restrict__ X,
                                                         float* __restrict__ ws) {
    int r = blockIdx.x * blockDim.x + threadIdx.x;
    if (r < MAT_N) {
        const float4* p = (const float4*)(X + r * MAT_D);
        float s = 0.0f;
#pragma unroll
        for (int i = 0; i < MAT_D / 4; ++i) {
            float4 v = p[i];
            s += v.x * v.x + v.y * v.y + v.z * v.z + v.w * v.w;
        }
        ws[r] = s;
    }
}

__global__ __launch_bounds__(256) void matern_setup_kernel(const float* __restrict__ nu_p,
                                                           const float* __restrict__ logvar_p,
                                                           const float* __restrict__ logl_p,
                                                           float* __restrict__ ws) {
    int tid = threadIdx.x;
    float nu = nu_p[0];
    if (tid < NQ) {
        const float h = 6.0f / (float)(NQ - 1);
        float t = h * (float)tid;
        float w = h;
        if (tid == 0 || tid == NQ - 1) w *= 0.5f;
        // Bake the negation and the exp->exp2 conversion factor into the table.
        ws[WS_TAB + 2 * tid + 0] = -(coshf(t) - 1.0f) * LOG2E_F;
        ws[WS_TAB + 2 * tid + 1] = w * coshf(nu * t);
    }
    if (tid == 0) {
        float var = expf(logvar_p[0]);
        float len = expf(logl_p[0]);
        ws[WS_PARAM + 0] = sqrtf(2.0f * nu) / len;
        ws[WS_PARAM + 1] = var;
        ws[WS_PARAM + 2] = ((1.0f - nu) * LN2_F - lgammaf(nu)) * LOG2E_F;
        ws[WS_PARAM + 3] = nu;
    }
}

// Fused kernel: one wave32 computes one 16x16 output tile.
//  - 8x V_WMMA_F32_16X16X4_F32 accumulate the Gram tile (K = 32).
//  - Epilogue: distance -> Matern via 192-pt quadrature -> store final value.
__global__ __launch_bounds__(256) void matern_main_kernel(const float* __restrict__ X,
                                                          const float* __restrict__ ws,
                                                          float* __restrict__ out) {
    __shared__ float2 s_tab[NQ];   // (c', w) pairs -> one ds_load_b64 per quad step

    {
        const float2* g_tab = (const float2*)(ws + WS_TAB);
        for (int i = threadIdx.x; i < NQ; i += 256) s_tab[i] = g_tab[i];
    }
    const float c_scale   = ws[WS_PARAM + 0];
    const float var       = ws[WS_PARAM + 1];
    const float log_norm2 = ws[WS_PARAM + 2];
    const float nu        = ws[WS_PARAM + 3];
    __syncthreads();

    const int wave = threadIdx.x >> 5;
    const int lane = threadIdx.x & 31;
    const int tile = blockIdx.x * 8 + wave;          // 512 blocks * 8 waves = 4096 tiles
    const int bi = tile >> 6;                        // 64 tiles per dimension
    const int bj = tile & 63;
    const int half = lane >> 4;                      // 0: lanes 0-15, 1: lanes 16-31
    const int lr = lane & 15;

    // A fragment (16x4 f32): VGPR0 = {K=0 | K=2}, VGPR1 = {K=1 | K=3} -> for
    // lane-half h we load columns 4s+2h, 4s+2h+1 of row (bi*16 + lr).
    // B = X^T, so the B fragment is the same gather from the column-tile rows.
    const float* Arow = X + (bi * 16 + lr) * MAT_D + 2 * half;
    const float* Brow = X + (bj * 16 + lr) * MAT_D + 2 * half;

    v8f acc = {};
#pragma unroll
    for (int s = 0; s < 8; ++s) {
        v2f a = *(const v2f*)(Arow + 4 * s);
        v2f b = *(const v2f*)(Brow + 4 * s);
        acc = __builtin_amdgcn_wmma_f32_16x16x4_f32(
            /*neg_a=*/false, a, /*neg_b=*/false, b,
            /*c_mod=*/(short)0, acc, /*reuse_a=*/false, /*reuse_b=*/false);
    }

    // C/D layout: VGPR r, lanes 0-15 -> M=r, lanes 16-31 -> M=r+8; N = lane%16.
    const int col = bj * 16 + lr;
    const float sqn_j = ws[col];

    float x[8];
#pragma unroll
    for (int r = 0; r < 8; ++r) {
        const int row = bi * 16 + 8 * half + r;
        float sq = ws[row] + sqn_j - 2.0f * acc[r];
        sq = fmaxf(sq, 1e-24f);
        float dist = __builtin_amdgcn_sqrtf(sq);     // raw v_sqrt_f32 (input is normal-range)
        x[r] = fmaxf(c_scale * dist, 1e-10f);
    }

    // K_nu(x) = exp(-x) * sum_q w_q cosh(nu t_q) exp(-x (cosh t_q - 1))
    //         = exp(-x) * sum_q w_q cosh(nu t_q) * 2^(x * c'_q)
    float integ[8] = {0.f, 0.f, 0.f, 0.f, 0.f, 0.f, 0.f, 0.f};
    for (int q = 0; q < NQ; ++q) {
        const float2 cw = s_tab[q];
#pragma unroll
        for (int r = 0; r < 8; ++r) {
            integ[r] = fmaf(cw.y, __builtin_amdgcn_exp2f(x[r] * cw.x), integ[r]);
        }
    }

#pragma unroll
    for (int r = 0; r < 8; ++r) {
        const int row = bi * 16 + 8 * half + r;
        // var * 2^(log_norm2 + nu*log2(x) - x*log2e) * integral
        float l2x  = __builtin_amdgcn_logf(x[r]);    // native v_log_f32 (log2)
        float e    = fmaf(nu, l2x, log_norm2) - x[r] * LOG2E_F;
        float val  = var * __builtin_amdgcn_exp2f(e) * integ[r];
        if (row == col) val = var;                   // diagonal is exactly var
        out[row * MAT_N + col] = val;
    }
}

extern "C" void kernel_launch(void* const* d_in, const int* in_sizes, int n_in,
                              void* d_out, int out_size, void* d_ws, size_t ws_size,
                              hipStream_t stream) {
    const float* X    = (const float*)d_in[0];
    const float* nu   = (const float*)d_in[1];
    const float* lvar = (const float*)d_in[2];
    const float* llen = (const float*)d_in[3];
    float* ws  = (float*)d_ws;
    float* out = (float*)d_out;

    matern_sqn_kernel<<<MAT_N / 256, 256, 0, stream>>>(X, ws);
    matern_setup_kernel<<<1, 256, 0, stream>>>(nu, lvar, llen, ws);
    matern_main_kernel<<<(64 * 64) / 8, 256, 0, stream>>>(X, ws, out);
}